// PoseLoss2D_80831284510873
// MI455X (gfx1250) — compile-verified
//
#include <hip/hip_runtime.h>

#define BINS   64
#define KPN    24
#define BATCH  256
#define HW     (BINS * BINS)
#define MIN_L  (-150.0f)
#define DELTA  (300.0f / 64.0f)
#define LOG2E  1.44269504088896340736f

typedef __attribute__((ext_vector_type(2))) float v2f;
typedef __attribute__((ext_vector_type(8))) float v8f;

__device__ __forceinline__ float center_of(int i) {
    return MIN_L + ((float)i + 0.5f) * DELTA;
}

// One block per batch element b. 768 threads: t = w0*24 + k.
// Thread (w0,k) accumulates online-softmax stats for columns w0 and w0+32 of
// keypoint k over all 64 rows h (one contiguous 6KB row read per iteration).
__global__ __launch_bounds__(768) void pose_stage1(
    const float* __restrict__ feat, const float* __restrict__ gt,
    float* __restrict__ out_loc, float* __restrict__ lossbuf)
{
    const int b  = blockIdx.x;
    const int t  = threadIdx.x;
    const int k  = t % KPN;   // keypoint
    const int w0 = t / KPN;   // column group 0..31 (owns w0 and w0+32)

    __shared__ float s_m[32 * KPN];    // per-thread running max (log2 domain)
    __shared__ float s_Mk[KPN];        // per-keypoint global max
    __shared__ float s_part[32 * 128]; // rows = w0, cols = q*24+k (q=0..4), padded to 128
    __shared__ float s_red[128];       // WMMA column sums

    const float* p = feat + (size_t)b * (HW * KPN) + (size_t)(w0 * KPN + k);

    float m = -1.0e30f;                 // running max of x*log2(e)
    float sA = 0.f, sB = 0.f;           // Sum exp per column (A=w0, B=w0+32)
    float sy = 0.f, sy2 = 0.f;          // y-moment sums (cy shared by both cols)

    #pragma unroll 4
    for (int h = 0; h < BINS; ++h) {
        float xa = p[h * (BINS * KPN)];
        float xb = p[h * (BINS * KPN) + 32 * KPN];
        float xsa = xa * LOG2E;
        float xsb = xb * LOG2E;
        float mo  = m;
        m = fmaxf(m, fmaxf(xsa, xsb));
        if (m > mo) {                    // rare; exp2(0)=1 for unchanged lanes
            float c = __builtin_amdgcn_exp2f(mo - m);
            sA *= c; sB *= c; sy *= c; sy2 *= c;
        }
        float ea = __builtin_amdgcn_exp2f(xsa - m);
        float eb = __builtin_amdgcn_exp2f(xsb - m);
        sA += ea;
        sB += eb;
        float es = ea + eb;
        float cy = MIN_L + ((float)h + 0.5f) * DELTA;
        sy  = fmaf(es, cy,      sy);
        sy2 = fmaf(es, cy * cy, sy2);
    }

    s_m[t] = m;
    __syncthreads();

    if (t < KPN) {                       // per-keypoint max over 32 partials
        float mm = s_m[t];
        #pragma unroll 4
        for (int j = 1; j < 32; ++j) mm = fmaxf(mm, s_m[j * KPN + t]);
        s_Mk[t] = mm;
    }
    __syncthreads();

    {   // rescale partials to the common max; defer x-weights to this stage
        float scale = __builtin_amdgcn_exp2f(m - s_Mk[k]);
        float a  = sA * scale, bb = sB * scale;
        float cxa = center_of(w0), cxb = center_of(w0 + 32);
        s_part[w0 * 128 + 0 * KPN + k] = a + bb;                        // S
        s_part[w0 * 128 + 1 * KPN + k] = sy  * scale;                   // Sy
        s_part[w0 * 128 + 2 * KPN + k] = sy2 * scale;                   // Sy2
        s_part[w0 * 128 + 3 * KPN + k] = a * cxa + bb * cxb;            // Sx
        s_part[w0 * 128 + 4 * KPN + k] = a * cxa * cxa + bb * cxb * cxb;// Sx2
    }
    // zero pad columns 120..127 so WMMA never sums garbage
    if (t < 256) s_part[(t >> 3) * 128 + 120 + (t & 7)] = 0.0f;
    __syncthreads();

    // WMMA column-sum over the 32 rows: D = Ones(16x4) x B(4x16) chained 8x.
    // Every row of D equals the column sums (exact in f32).
    const int wid  = t >> 5;
    const int lane = t & 31;
    if (wid < 8) {
        v2f ones = {1.0f, 1.0f};
        v8f acc  = {};
        int col  = wid * 16 + (lane & 15);
        #pragma unroll
        for (int c8 = 0; c8 < 8; ++c8) {
            int rbase = c8 * 4;
            v2f bvec;
            if (lane < 16) {            // lanes 0-15: K rows 0,1 of this step
                bvec.x = s_part[(rbase + 0) * 128 + col];
                bvec.y = s_part[(rbase + 1) * 128 + col];
            } else {                    // lanes 16-31: K rows 2,3
                bvec.x = s_part[(rbase + 2) * 128 + col];
                bvec.y = s_part[(rbase + 3) * 128 + col];
            }
            acc = __builtin_amdgcn_wmma_f32_16x16x4_f32(
                false, ones, false, bvec, (short)0, acc, false, false);
        }
        if (lane < 16) s_red[col] = acc[0];   // D[M=0][N=lane]
    }
    __syncthreads();

    if (t < KPN) {
        float S   = s_red[0 * KPN + t];
        float Sy  = s_red[1 * KPN + t];
        float Sy2 = s_red[2 * KPN + t];
        float Sx  = s_red[3 * KPN + t];
        float Sx2 = s_red[4 * KPN + t];
        float inv = 1.0f / S;
        float ly = Sy * inv, lx = Sx * inv;
        float vy = Sy2 * inv - ly * ly;
        float vx = Sx2 * inv - lx * lx;
        float gy = gt[(size_t)b * KPN * 2 + t * 2 + 0];
        float gx = gt[(size_t)b * KPN * 2 + t * 2 + 1];
        float dy = ly - gy, dx = lx - gx;
        lossbuf[(size_t)b * KPN * 2 + t * 2 + 0] = vy + dy * dy;
        lossbuf[(size_t)b * KPN * 2 + t * 2 + 1] = vx + dx * dx;
        out_loc[(size_t)b * KPN * 2 + t * 2 + 0] = ly;
        out_loc[(size_t)b * KPN * 2 + t * 2 + 1] = lx;
    }
}

// Single block: global loss_sum reduction + transposed hard-loss epilogue.
__global__ __launch_bounds__(1024) void pose_stage2(
    const float* __restrict__ lossbuf, float* __restrict__ out_hard)
{
    __shared__ float red[1024];
    __shared__ float s_inv;
    const int t = threadIdx.x;
    const int N = 2 * KPN * BATCH;   // 12288

    float s = 0.f;
    for (int i = t; i < N; i += 1024) s += lossbuf[i];
    red[t] = s;
    __syncthreads();
    #pragma unroll
    for (int off = 512; off > 0; off >>= 1) {
        if (t < off) red[t] += red[t + off];
        __syncthreads();
    }
    if (t == 0) {
        float loss_sum = red[0] + 0.001f;
        s_inv = ((float)KPN / 150.0f) / loss_sum;   // kp_f / loss_sum / MAX_LOC
    }
    __syncthreads();
    float inv = s_inv;

    for (int o = t; o < N; o += 1024) {             // out layout: [2, KP, B]
        int c = o / (KPN * BATCH);
        int r = o % (KPN * BATCH);
        int k = r / BATCH;
        int b = r % BATCH;
        float l  = lossbuf[b * KPN * 2 + k * 2 + c];
        float l2 = lossbuf[b * KPN * 2 + k * 2 + 0] +
                   lossbuf[b * KPN * 2 + k * 2 + 1];
        out_hard[o] = l * l2 * inv;
    }
}

extern "C" void kernel_launch(void* const* d_in, const int* in_sizes, int n_in,
                              void* d_out, int out_size, void* d_ws, size_t ws_size,
                              hipStream_t stream) {
    const float* feat = (const float*)d_in[0];   // [256, 64, 64, 24] f32
    const float* gt   = (const float*)d_in[1];   // [256, 24, 2]      f32
    float* out      = (float*)d_out;
    float* out_hard = out;                        // [2, 24, 256] -> 12288
    float* out_loc  = out + 2 * KPN * BATCH;      // [256, 24, 2] -> 12288
    float* lossbuf  = (float*)d_ws;               // 12288 f32 scratch

    pose_stage1<<<BATCH, 768, 0, stream>>>(feat, gt, out_loc, lossbuf);
    pose_stage2<<<1, 1024, 0, stream>>>(lossbuf, out_hard);
}